// VGAE_48808008351905
// MI455X (gfx1250) — compile-verified
//
#include <hip/hip_runtime.h>

#define NNODES 100000
#define NFEAT  512
#define HID    16
#define NEDGES 3200000

typedef float v2f __attribute__((ext_vector_type(2)));
typedef float v8f __attribute__((ext_vector_type(8)));

__device__ __forceinline__ void atomic_add_f32(float* p, float v) {
    // relaxed, device scope -> no-return global_atomic_add_f32
    __hip_atomic_fetch_add(p, v, __ATOMIC_RELAXED, __HIP_MEMORY_SCOPE_AGENT);
}

// ---------------- degree / normalization ----------------

__global__ void k_deg_init(float* __restrict__ deg) {
    int i = blockIdx.x * blockDim.x + threadIdx.x;
    if (i < NNODES) deg[i] = 1.0f;  // self-loop contribution
}

__global__ void k_deg_count(const long long* __restrict__ dst, float* __restrict__ deg) {
    int e = blockIdx.x * blockDim.x + threadIdx.x;
    if (e < NEDGES) atomic_add_f32(&deg[(int)dst[e]], 1.0f);
}

__global__ void k_rsqrt(float* __restrict__ deg) {
    int i = blockIdx.x * blockDim.x + threadIdx.x;
    if (i < NNODES) deg[i] = rsqrtf(deg[i]);
}

// ---------------- GEMM1: h = x @ W1  (M=100000, K=512, N=16), f32 WMMA ----------------
// One wave32 per 16x16 output tile. A: 16x4 f32 (2 VGPRs/lane), B: 4x16 (2 VGPRs/lane).
// A lane layout: lanes 0-15 row=lane, K={k,k+1}; lanes 16-31 row=lane-16, K={k+2,k+3}.
// B lane layout (mirrored): lanes 0-15 col=lane, K={k,k+1} in v0,v1; lanes 16-31 K={k+2,k+3}.
// C/D: VGPR j -> lanes 0-15: M=j, lanes 16-31: M=8+j; N=lane&15.

__global__ __launch_bounds__(128) void k_gemm1(const float* __restrict__ x,
                                               const float* __restrict__ W,
                                               float* __restrict__ h) {
    __shared__ float Ws[NFEAT * HID];  // 32 KB of 320 KB WGP LDS
    for (int i = threadIdx.x; i < NFEAT * HID; i += blockDim.x) Ws[i] = W[i];
    __syncthreads();

    const int wave = blockIdx.x * (blockDim.x >> 5) + (threadIdx.x >> 5);
    const int lane = threadIdx.x & 31;
    const int nTiles = NNODES / 16;     // 6250, exact
    if (wave >= nTiles) return;         // wave-uniform: EXEC stays all-ones inside

    const int row = lane & 15;
    const int kh  = lane >> 4;          // which K-pair this half-wave holds
    const int col = lane & 15;
    const float* __restrict__ xrow = x + ((size_t)(wave * 16 + row)) * NFEAT;

    v8f c = {};
    for (int k = 0; k < NFEAT; k += 4) {
        __builtin_prefetch(xrow + k + 64, 0, 0);       // global_prefetch_b8, stream x
        v2f a = *(const v2f*)(xrow + k + 2 * kh);      // 8B-aligned
        v2f b;
        b.x = Ws[(k + 2 * kh + 0) * HID + col];
        b.y = Ws[(k + 2 * kh + 1) * HID + col];
        c = __builtin_amdgcn_wmma_f32_16x16x4_f32(false, a, false, b,
                                                  (short)0, c, false, false);
    }

    float* __restrict__ hrow = h + (size_t)wave * 16 * HID;
    for (int j = 0; j < 8; ++j) {
        int m = kh * 8 + j;
        hrow[m * HID + col] = c[j];
    }
}

// ---------------- GEMM2: g = h2 @ W2  (M=100000, K=16, N=16) ----------------

__global__ __launch_bounds__(128) void k_gemm2(const float* __restrict__ hin,
                                               const float* __restrict__ W,
                                               float* __restrict__ g) {
    __shared__ float Ws[HID * HID];
    if (threadIdx.x < HID * HID) Ws[threadIdx.x] = W[threadIdx.x];
    __syncthreads();

    const int wave = blockIdx.x * (blockDim.x >> 5) + (threadIdx.x >> 5);
    const int lane = threadIdx.x & 31;
    const int nTiles = NNODES / 16;
    if (wave >= nTiles) return;

    const int row = lane & 15;
    const int kh  = lane >> 4;
    const int col = lane & 15;
    const float* __restrict__ arow = hin + ((size_t)(wave * 16 + row)) * HID;

    v8f c = {};
    for (int k = 0; k < HID; k += 4) {
        v2f a = *(const v2f*)(arow + k + 2 * kh);
        v2f b;
        b.x = Ws[(k + 2 * kh + 0) * HID + col];
        b.y = Ws[(k + 2 * kh + 1) * HID + col];
        c = __builtin_amdgcn_wmma_f32_16x16x4_f32(false, a, false, b,
                                                  (short)0, c, false, false);
    }

    float* __restrict__ grow = g + (size_t)wave * 16 * HID;
    for (int j = 0; j < 8; ++j) {
        int m = kh * 8 + j;
        grow[m * HID + col] = c[j];
    }
}

// ---------------- segment-sum init (self-loop + bias) ----------------
// acc[i,f] = bias[f] + hh[i,f] * dis[i]^2   (self-loop norm = dis[i]*dis[i])

__global__ void k_seg_init(const float* __restrict__ hh, const float* __restrict__ dis,
                           const float* __restrict__ bias, float* __restrict__ acc) {
    int t = blockIdx.x * blockDim.x + threadIdx.x;
    if (t < NNODES * HID) {
        int i = t >> 4, f = t & 15;
        float d = dis[i];
        acc[t] = bias[f] + hh[t] * d * d;
    }
}

// ---------------- edge scatter: acc[d,:] += h[s,:] * dis[s]*dis[d] ----------------
// 16 lanes per edge -> 16 consecutive f32 atomics per edge (one cacheline).

__global__ void k_scatter(const long long* __restrict__ src, const long long* __restrict__ dst,
                          const float* __restrict__ hh, const float* __restrict__ dis,
                          float* __restrict__ acc) {
    int t = blockIdx.x * blockDim.x + threadIdx.x;
    if (t >= NEDGES * HID) return;
    int e = t >> 4, f = t & 15;
    int s = (int)src[e];
    int d = (int)dst[e];
    float w = dis[s] * dis[d];
    atomic_add_f32(&acc[(size_t)d * HID + f], hh[(size_t)s * HID + f] * w);
}

// ---------------- elementwise ----------------

__global__ void k_relu(const float* __restrict__ in, float* __restrict__ out) {
    int t = blockIdx.x * blockDim.x + threadIdx.x;
    if (t < NNODES * HID) out[t] = fmaxf(in[t], 0.0f);
}

__global__ void k_final(const float* __restrict__ eps, float* __restrict__ z) {
    int t = blockIdx.x * blockDim.x + threadIdx.x;
    if (t < NNODES * HID) {
        float v = z[t];
        z[t] = v + eps[t] * __expf(0.5f * v);   // v_exp_f32
    }
}

// ---------------- launch ----------------

extern "C" void kernel_launch(void* const* d_in, const int* in_sizes, int n_in,
                              void* d_out, int out_size, void* d_ws, size_t ws_size,
                              hipStream_t stream) {
    const float*     x   = (const float*)d_in[0];
    const float*     W1  = (const float*)d_in[1];
    const float*     b1  = (const float*)d_in[2];
    const float*     W2  = (const float*)d_in[3];
    const float*     b2  = (const float*)d_in[4];
    const float*     eps = (const float*)d_in[5];
    const long long* ei  = (const long long*)d_in[6];
    const long long* src = ei;
    const long long* dst = ei + NEDGES;

    float* ws  = (float*)d_ws;
    float* dis = ws;                              // N floats (deg -> rsqrt in place)
    float* h   = ws + NNODES;                     // 16N floats
    float* acc = h + (size_t)NNODES * HID;        // 16N floats
    float* out = (float*)d_out;                   // N x 16 f32

    const int tiles  = NNODES / 16;               // 6250
    const int gemmBk = (tiles + 3) / 4;           // 4 waves / 128 threads per block
    const int nodeT  = (NNODES * HID + 255) / 256;
    const int edgeT  = (NEDGES + 255) / 256;
    const int edgeFT = (NEDGES * HID + 255) / 256;

    // normalization
    k_deg_init <<<(NNODES + 255) / 256, 256, 0, stream>>>(dis);
    k_deg_count<<<edgeT, 256, 0, stream>>>(dst, dis);
    k_rsqrt    <<<(NNODES + 255) / 256, 256, 0, stream>>>(dis);

    // layer 1: h = x@W1 ; acc = b1 + self-loop + edge messages ; h = relu(acc)
    k_gemm1   <<<gemmBk, 128, 0, stream>>>(x, W1, h);
    k_seg_init<<<nodeT, 256, 0, stream>>>(h, dis, b1, acc);
    k_scatter <<<edgeFT, 256, 0, stream>>>(src, dst, h, dis, acc);
    k_relu    <<<nodeT, 256, 0, stream>>>(acc, h);

    // layer 2: acc = h@W2 ; out = b2 + self-loop + edge messages
    k_gemm2   <<<gemmBk, 128, 0, stream>>>(h, W2, acc);
    k_seg_init<<<nodeT, 256, 0, stream>>>(acc, dis, b2, out);
    k_scatter <<<edgeFT, 256, 0, stream>>>(src, dst, acc, dis, out);

    // reparameterize: out = z + eps*exp(0.5 z)
    k_final   <<<nodeT, 256, 0, stream>>>(eps, out);
}